// KalmanVAE_33079838114281
// MI455X (gfx1250) — compile-verified
//
#include <hip/hip_runtime.h>

// Problem dims (compile-time constants from the reference)
#define B_ 64
#define T_ 100
#define D_ 32
#define L_ 64
#define K_ 16
#define H_ 50

typedef __attribute__((ext_vector_type(2))) float v2f;
typedef __attribute__((ext_vector_type(8))) float v8f;

// V_WMMA_F32_16X16X4_F32: D(16x16 f32) = A(16x4) * B(4x16) + C
__device__ __forceinline__ v8f wmma4(v2f a, v2f b, v8f c) {
  return __builtin_amdgcn_wmma_f32_16x16x4_f32(
      false, a, false, b, (short)0, c, false, false);
}

// ---------------------------------------------------------------------------
// Async LDS <-> global DMA (gfx1250). Tracked by ASYNCcnt.
//   load : copies 16 bytes per lane from the global address in vaddr to the LDS byte address in vdst
//   store: copies 16 bytes per lane from the LDS byte address in vsrc to the global address in vaddr
// LDS byte address = low 32 bits of the generic shared-space pointer.
// ---------------------------------------------------------------------------
__device__ __forceinline__ void async_ld_b128(uint32_t lds_byte, const void* gp) {
  asm volatile("global_load_async_to_lds_b128 %0, %1, off"
               :: "v"(lds_byte), "v"(gp) : "memory");
}
__device__ __forceinline__ void async_st_b128(void* gp, uint32_t lds_byte) {
  asm volatile("global_store_async_from_lds_b128 %0, %1, off"
               :: "v"(gp), "v"(lds_byte) : "memory");
}
__device__ __forceinline__ void wait_async0() {
  asm volatile("s_wait_asynccnt 0" ::: "memory");
}
__device__ __forceinline__ uint32_t lds_addr(const void* p) {
  return (uint32_t)(uintptr_t)p;
}

// Workgroup-cooperative LDS matmul: C = alpha*(opA(A) @ opB(B)) + beta*C
// M,N multiples of 16; K multiple of 4. 256 threads = 8 wave32; one 16x16
// output tile per wave per trip. Barriers at entry/exit.
template <int M, int N, int K, bool TA, bool TB>
__device__ __forceinline__ void wg_mm(const float* A, int lda,
                                      const float* Bm, int ldb,
                                      float* C, int ldc,
                                      float alpha, float beta) {
  __syncthreads();
  const int lane = threadIdx.x & 31;
  const int wid  = threadIdx.x >> 5;
  constexpr int MT = M / 16, NT = N / 16;
  const int l15 = lane & 15;
  const int k0  = (lane < 16) ? 0 : 2;
  for (int tile = wid; tile < MT * NT; tile += 8) {
    const int tm = tile / NT, tn = tile % NT;
    const int ra   = tm * 16 + l15;                    // A row
    const int cb   = tn * 16 + l15;                    // B col
    const int crow = tm * 16 + ((lane < 16) ? 0 : 8);  // C base row
    const int ccol = tn * 16 + l15;                    // C col
    v8f acc;
    if (beta != 0.0f) {
      const float s = beta / alpha;
#pragma unroll
      for (int r = 0; r < 8; ++r) acc[r] = s * C[(crow + r) * ldc + ccol];
    } else {
#pragma unroll
      for (int r = 0; r < 8; ++r) acc[r] = 0.0f;
    }
#pragma unroll
    for (int k = 0; k < K; k += 4) {
      const int ka = k + k0;
      v2f av, bv;
      av.x = TA ? A[ka * lda + ra]       : A[ra * lda + ka];
      av.y = TA ? A[(ka + 1) * lda + ra] : A[ra * lda + ka + 1];
      bv.x = TB ? Bm[cb * ldb + ka]      : Bm[ka * ldb + cb];
      bv.y = TB ? Bm[cb * ldb + ka + 1]  : Bm[(ka + 1) * ldb + cb];
      acc = wmma4(av, bv, acc);
    }
#pragma unroll
    for (int r = 0; r < 8; ++r) C[(crow + r) * ldc + ccol] = alpha * acc[r];
  }
  __syncthreads();
}

// Workgroup-cooperative Gauss-Jordan inverse (no pivoting; SPD inputs).
template <int NN>
__device__ __forceinline__ void wg_inv(const float* Min, float* Mout,
                                       float* aug, float* fac) {
  constexpr int W = 2 * NN;
  for (int idx = threadIdx.x; idx < NN * W; idx += blockDim.x) {
    const int r = idx / W, c = idx - r * W;
    aug[idx] = (c < NN) ? Min[r * NN + c] : ((c - NN) == r ? 1.0f : 0.0f);
  }
  __syncthreads();
  for (int p = 0; p < NN; ++p) {
    if (threadIdx.x < NN) fac[threadIdx.x] = aug[threadIdx.x * W + p];
    __syncthreads();
    const float ipv = 1.0f / fac[p];
    for (int c = threadIdx.x; c < W; c += blockDim.x) aug[p * W + c] *= ipv;
    __syncthreads();
    for (int idx = threadIdx.x; idx < NN * W; idx += blockDim.x) {
      const int r = idx / W, c = idx - r * W;
      if (r != p) aug[idx] -= fac[r] * aug[p * W + c];
    }
    __syncthreads();
  }
  for (int idx = threadIdx.x; idx < NN * NN; idx += blockDim.x) {
    const int r = idx / NN, c = idx - r * NN;
    Mout[idx] = aug[r * W + NN + c];
  }
  __syncthreads();
}

// Kernel 1: MLP -> softmax mode weights w (B,T,K). One thread per (b,t).
__global__ void mlp_weights_kernel(const float* __restrict__ obs,
                                   const float* __restrict__ start_code,
                                   const float* __restrict__ W1,
                                   const float* __restrict__ b1,
                                   const float* __restrict__ W2,
                                   const float* __restrict__ b2,
                                   float* __restrict__ w) {
  const int m = blockIdx.x * blockDim.x + threadIdx.x;
  if (m >= B_ * T_) return;
  const int b = m / T_, t = m - b * T_;
  const float* x = (t == 0) ? start_code : (obs + (size_t)(b * T_ + (t - 1)) * D_);
  float h[H_];
#pragma unroll 1
  for (int j = 0; j < H_; ++j) {
    float s = b1[j];
    for (int d = 0; d < D_; ++d) s += x[d] * W1[d * H_ + j];
    h[j] = s > 0.0f ? s : 0.0f;
  }
  float e[K_];
  float mx = -3.4e38f;
#pragma unroll 1
  for (int k = 0; k < K_; ++k) {
    float s = b2[k];
    for (int j = 0; j < H_; ++j) s += h[j] * W2[j * K_ + k];
    e[k] = s;
    mx = fmaxf(mx, s);
  }
  float sum = 0.0f;
  for (int k = 0; k < K_; ++k) { e[k] = __expf(e[k] - mx); sum += e[k]; }
  const float inv = 1.0f / sum;
  for (int k = 0; k < K_; ++k) w[(size_t)m * K_ + k] = e[k] * inv;
}

// Kernel 2: mode-mixing GEMM (6400x16)x(16x6144) via WMMA; n<4096 -> A_t,
// else C_t. One 16x16 tile per wave, K=16 in 4 chained wmma steps.
__global__ void __launch_bounds__(256)
mix_kernel(const float* __restrict__ w, const float* __restrict__ A,
           const float* __restrict__ C, float* __restrict__ At,
           float* __restrict__ Ct) {
  const int lane   = threadIdx.x & 31;
  const int gwid   = (blockIdx.x * blockDim.x + threadIdx.x) >> 5;
  const int nwaves = (gridDim.x * blockDim.x) >> 5;
  constexpr int NA = L_ * L_;         // 4096
  constexpr int NC = D_ * L_;         // 2048
  constexpr int MT = (B_ * T_) / 16;  // 400
  constexpr int NT = (NA + NC) / 16;  // 384
  const int l15 = lane & 15;
  const int k0  = (lane < 16) ? 0 : 2;
  for (int tile = gwid; tile < MT * NT; tile += nwaves) {
    const int tm = tile / NT, tn = tile % NT;
    const int m = tm * 16 + l15;
    const int n = tn * 16 + l15;
    const bool inA = (tn * 16) < NA;
    const float* Bsrc = inA ? A : C;
    const int ldb = inA ? NA : NC;
    const int nb  = inA ? n : (n - NA);
    v8f acc;
#pragma unroll
    for (int r = 0; r < 8; ++r) acc[r] = 0.0f;
#pragma unroll
    for (int k = 0; k < K_; k += 4) {
      const int ka = k + k0;
      v2f av, bv;
      av.x = w[(size_t)m * K_ + ka];
      av.y = w[(size_t)m * K_ + ka + 1];
      bv.x = Bsrc[(size_t)ka * ldb + nb];
      bv.y = Bsrc[(size_t)(ka + 1) * ldb + nb];
      acc = wmma4(av, bv, acc);
    }
    const int crow = tm * 16 + ((lane < 16) ? 0 : 8);
#pragma unroll
    for (int r = 0; r < 8; ++r) {
      const int row = crow + r;
      if (inA) At[(size_t)row * NA + n]        = acc[r];
      else     Ct[(size_t)row * NC + (n - NA)] = acc[r];
    }
  }
}

// Kernel 3: Kalman forward filter, one workgroup per batch element.
// Bulk LDS staging and matrix write-back go through the async DMA path.
__global__ void __launch_bounds__(256)
kalman_filter_kernel(const float* __restrict__ obs,
                     const float* __restrict__ At_g,
                     const float* __restrict__ Ct_g,
                     float* __restrict__ mu_f, float* __restrict__ Sig_f,
                     float* __restrict__ mu_p, float* __restrict__ Sig_p) {
  extern __shared__ float sm[];
  float* Sig  = sm;             // 4096
  float* Amat = Sig  + 4096;    // 4096
  float* Im   = Amat + 4096;    // 4096
  float* T1   = Im   + 4096;    // 4096
  float* SigZ = T1   + 4096;    // 4096
  float* Cm   = SigZ + 4096;    // 2048
  float* CS   = Cm   + 2048;    // 2048
  float* Kg   = CS   + 2048;    // 2048
  float* SCT  = Kg   + 2048;    // 2048
  float* aug  = SCT  + 2048;    // 2048 (32x64)
  float* S    = aug  + 2048;    // 1024
  float* Sinv = S    + 1024;    // 1024
  float* mu   = Sinv + 1024;    // 64
  float* muz  = mu   + 64;      // 64
  float* rv   = muz  + 64;      // 64
  float* fac  = rv   + 64;      // 64
  float* yv   = fac  + 64;      // 64

  const int b   = blockIdx.x;
  const int tid = threadIdx.x;

  for (int i = tid; i < L_ * L_; i += blockDim.x)
    Sig[i] = ((i / L_) == (i % L_)) ? 20.0f : 0.0f;
  if (tid < L_) mu[tid] = 0.0f;
  __syncthreads();

  for (int t = 0; t < T_; ++t) {
    const float* Ag = At_g + (size_t)(b * T_ + t) * (L_ * L_);
    const float* Cg = Ct_g + (size_t)(b * T_ + t) * (D_ * L_);
    if (t + 1 < T_) {
      __builtin_prefetch(At_g + (size_t)(b * T_ + t + 1) * (L_ * L_), 0, 1);
      __builtin_prefetch(Ct_g + (size_t)(b * T_ + t + 1) * (D_ * L_), 0, 1);
    }
    // async DMA: stage A_t, C_t into LDS; stream Sig prior out to workspace
    for (int i = tid; i < (L_ * L_) / 4; i += blockDim.x)
      async_ld_b128(lds_addr(Amat) + i * 16, Ag + i * 4);
    for (int i = tid; i < (D_ * L_) / 4; i += blockDim.x)
      async_ld_b128(lds_addr(Cm) + i * 16, Cg + i * 4);
    for (int i = tid; i < (L_ * L_) / 4; i += blockDim.x)
      async_st_b128(Sig_p + (size_t)(t * B_ + b) * (L_ * L_) + i * 4,
                    lds_addr(Sig) + i * 16);
    if (tid < D_) yv[tid] = obs[(size_t)(b * T_ + t) * D_ + tid];
    if (tid < L_) mu_p[(size_t)(t * B_ + b) * L_ + tid] = mu[tid];
    wait_async0();  // Amat/Cm ready (and prior DMA done before Sig is reused)
    __syncthreads();

    if (tid < D_) {  // r = y - C mu
      float s = 0.0f;
      for (int l = 0; l < L_; ++l) s += Cm[tid * L_ + l] * mu[l];
      rv[tid] = yv[tid] - s;
    }
    wg_mm<D_, L_, L_, false, false>(Cm, L_, Sig, L_, CS, L_, 1.0f, 0.0f);
    wg_mm<D_, D_, L_, false, true >(CS, L_, Cm,  L_, S,  D_, 1.0f, 0.0f);
    if (tid < D_) S[tid * D_ + tid] += 0.3f;
    __syncthreads();
    wg_inv<D_>(S, Sinv, aug, fac);
    wg_mm<L_, D_, L_, false, true >(Sig, L_, Cm,   L_, SCT, D_, 1.0f, 0.0f);
    wg_mm<L_, D_, D_, false, false>(SCT, D_, Sinv, D_, Kg,  D_, 1.0f, 0.0f);
    if (tid < L_) {  // mu_z = mu + Kg r
      float s = 0.0f;
      for (int d = 0; d < D_; ++d) s += Kg[tid * D_ + d] * rv[d];
      muz[tid] = mu[tid] + s;
    }
    wg_mm<L_, L_, D_, false, false>(Kg, D_, Cm, L_, Im, L_, -1.0f, 0.0f);
    if (tid < L_) Im[tid * L_ + tid] += 1.0f;
    __syncthreads();
    wg_mm<L_, L_, L_, false, false>(Im, L_, Sig, L_, T1,   L_, 1.0f, 0.0f);
    wg_mm<L_, L_, L_, false, true >(T1, L_, Im,  L_, SigZ, L_, 1.0f, 0.0f);
    wg_mm<L_, L_, D_, false, true >(Kg, D_, Kg,  D_, SigZ, L_, 0.3f, 1.0f);
    // async DMA: filtered covariance out (drained by next step's wait before
    // SigZ is overwritten)
    for (int i = tid; i < (L_ * L_) / 4; i += blockDim.x)
      async_st_b128(Sig_f + (size_t)(t * B_ + b) * (L_ * L_) + i * 4,
                    lds_addr(SigZ) + i * 16);
    if (tid < L_) mu_f[(size_t)(t * B_ + b) * L_ + tid] = muz[tid];
    __syncthreads();
    if (tid < L_) {  // mu = A mu_z
      float s = 0.0f;
      for (int l = 0; l < L_; ++l) s += Amat[tid * L_ + l] * muz[l];
      mu[tid] = s;
    }
    wg_mm<L_, L_, L_, false, false>(Amat, L_, SigZ, L_, T1,  L_, 1.0f, 0.0f);
    wg_mm<L_, L_, L_, false, true >(T1,   L_, Amat, L_, Sig, L_, 1.0f, 0.0f);
    if (tid < L_) Sig[tid * L_ + tid] += 0.8f;
    __syncthreads();
  }
  wait_async0();  // drain last filtered-covariance DMA before wave exit
}

// Kernel 4: RTS smoother (reverse, in-place over filtered outputs).
__global__ void __launch_bounds__(256)
rts_smoother_kernel(const float* __restrict__ At_g,
                    float* __restrict__ mu_sm, float* __restrict__ Sig_sm,
                    const float* __restrict__ mu_p,
                    const float* __restrict__ Sig_p) {
  extern __shared__ float sm[];
  float* SigS = sm;             // 4096
  float* Sf   = SigS + 4096;    // 4096
  float* Sp   = Sf   + 4096;    // 4096 (becomes Dif in place)
  float* A1   = Sp   + 4096;    // 4096
  float* SpI  = A1   + 4096;    // 4096
  float* J    = SpI  + 4096;    // 4096
  float* T1   = J    + 4096;    // 4096
  float* aug  = T1   + 4096;    // 8192 (64x128)
  float* muS  = aug  + 8192;    // 64
  float* mf   = muS  + 64;      // 64
  float* mp   = mf   + 64;      // 64
  float* fac  = mp   + 64;      // 64

  const int b   = blockIdx.x;
  const int tid = threadIdx.x;

  for (int i = tid; i < L_ * L_; i += blockDim.x)
    SigS[i] = Sig_sm[(size_t)((T_ - 1) * B_ + b) * (L_ * L_) + i];
  if (tid < L_) muS[tid] = mu_sm[(size_t)((T_ - 1) * B_ + b) * L_ + tid];
  __syncthreads();

  for (int t = T_ - 2; t >= 0; --t) {
    // async DMA: stage Sf(t), A(t+1), Sig_p(t+1)
    for (int i = tid; i < (L_ * L_) / 4; i += blockDim.x) {
      async_ld_b128(lds_addr(Sf) + i * 16,
                    Sig_sm + (size_t)(t * B_ + b) * (L_ * L_) + i * 4);
      async_ld_b128(lds_addr(A1) + i * 16,
                    At_g + (size_t)(b * T_ + (t + 1)) * (L_ * L_) + i * 4);
      async_ld_b128(lds_addr(Sp) + i * 16,
                    Sig_p + (size_t)((t + 1) * B_ + b) * (L_ * L_) + i * 4);
    }
    if (tid < L_) {
      mf[tid] = mu_sm[(size_t)(t * B_ + b) * L_ + tid];
      mp[tid] = mu_p[(size_t)((t + 1) * B_ + b) * L_ + tid];
    }
    wait_async0();  // loads ready; also drains previous step's SigS DMA
    __syncthreads();
    wg_inv<L_>(Sp, SpI, aug, fac);  // Sp not modified
    wg_mm<L_, L_, L_, false, true >(Sf, L_, A1,  L_, T1, L_, 1.0f, 0.0f);
    wg_mm<L_, L_, L_, false, false>(T1, L_, SpI, L_, J,  L_, 1.0f, 0.0f);
    if (tid < L_) mp[tid] = muS[tid] - mp[tid];
    for (int i = tid; i < L_ * L_; i += blockDim.x) Sp[i] = SigS[i] - Sp[i];
    __syncthreads();
    if (tid < L_) {
      float s = 0.0f;
      for (int l = 0; l < L_; ++l) s += J[tid * L_ + l] * mp[l];
      muS[tid] = mf[tid] + s;
    }
    wg_mm<L_, L_, L_, false, false>(J, L_, Sp, L_, T1, L_, 1.0f, 0.0f);
    for (int i = tid; i < L_ * L_; i += blockDim.x) SigS[i] = Sf[i];
    __syncthreads();
    wg_mm<L_, L_, L_, false, true >(T1, L_, J, L_, SigS, L_, 1.0f, 1.0f);
    // async DMA: smoothed covariance out (drained by next step's wait before
    // SigS is overwritten)
    for (int i = tid; i < (L_ * L_) / 4; i += blockDim.x)
      async_st_b128(Sig_sm + (size_t)(t * B_ + b) * (L_ * L_) + i * 4,
                    lds_addr(SigS) + i * 16);
    if (tid < L_) mu_sm[(size_t)(t * B_ + b) * L_ + tid] = muS[tid];
    __syncthreads();
  }
  wait_async0();  // drain final smoothed-covariance DMA before wave exit
}

extern "C" void kernel_launch(void* const* d_in, const int* in_sizes, int n_in,
                              void* d_out, int out_size, void* d_ws,
                              size_t ws_size, hipStream_t stream) {
  (void)in_sizes; (void)n_in; (void)out_size; (void)ws_size;
  const float* obs        = (const float*)d_in[0];
  const float* start_code = (const float*)d_in[1];
  const float* W1         = (const float*)d_in[2];
  const float* b1         = (const float*)d_in[3];
  const float* W2         = (const float*)d_in[4];
  const float* b2         = (const float*)d_in[5];
  const float* A          = (const float*)d_in[6];
  const float* C          = (const float*)d_in[7];

  float* out    = (float*)d_out;                      // tuple, concatenated
  float* mu_sm  = out;                                // (T,B,L)
  float* Sig_sm = mu_sm  + (size_t)T_ * B_ * L_;      // (T,B,L,L)
  float* At_o   = Sig_sm + (size_t)T_ * B_ * L_ * L_; // (B,T,L,L)
  float* Ct_o   = At_o   + (size_t)B_ * T_ * L_ * L_; // (B,T,D,L)

  float* ws    = (float*)d_ws;
  float* w     = ws;                                  // (B,T,K)
  float* mu_p  = w    + (size_t)B_ * T_ * K_;         // (T,B,L)
  float* Sig_p = mu_p + (size_t)T_ * B_ * L_;         // (T,B,L,L)

  mlp_weights_kernel<<<(B_ * T_ + 127) / 128, 128, 0, stream>>>(
      obs, start_code, W1, b1, W2, b2, w);
  mix_kernel<<<1200, 256, 0, stream>>>(w, A, C, At_o, Ct_o);

  const size_t smem_f =
      (size_t)(5 * 4096 + 5 * 2048 + 2 * 1024 + 5 * 64) * sizeof(float);
  kalman_filter_kernel<<<B_, 256, smem_f, stream>>>(obs, At_o, Ct_o, mu_sm,
                                                    Sig_sm, mu_p, Sig_p);
  const size_t smem_s = (size_t)(7 * 4096 + 8192 + 4 * 64) * sizeof(float);
  rts_smoother_kernel<<<B_, 256, smem_s, stream>>>(At_o, mu_sm, Sig_sm, mu_p,
                                                   Sig_p);
}